// DifferentiableLindblad_44813688766494
// MI455X (gfx1250) — compile-verified
//
#include <hip/hip_runtime.h>
#include <stdint.h>

// Liouvillian for two driven-dissipative Rydberg atoms.
// Output: (B,16,16) complex128 stored as interleaved double pairs.
// Re L = constant DECAY pattern, Im L = -(H (x) I) + (I (x) H^T), H real 4x4.

#define NTHREADS 256
static constexpr double kGamma = 1.0 / 88e-6;  // 1 / tau_eff

typedef int v4i __attribute__((vector_size(16)));
typedef v4i __attribute__((address_space(1))) *gv4i_ptr;   // global (device) AS
typedef v4i __attribute__((address_space(3))) *lv4i_ptr;   // LDS AS

__device__ __forceinline__ double Hval(int a, int b, double D1, double D2,
                                       double V, double hOm) {
    if (a == b) {
        if (a == 1) return -D2;
        if (a == 2) return -D1;
        if (a == 3) return (-D1 - D2) + V;
        return 0.0;
    }
    // off-diagonal: 0.5*Om on (0,1),(0,2),(1,3),(2,3) pairs; 0 on anti-diagonal
    return ((a ^ b) == 3) ? 0.0 : hOm;
}

__global__ __launch_bounds__(NTHREADS)
void lindblad_liouvillian_kernel(const float* __restrict__ Omega,
                                 const float* __restrict__ Delta,
                                 const float* __restrict__ dd1,
                                 const float* __restrict__ dd2,
                                 const float* __restrict__ dph,
                                 const float* __restrict__ Vv,
                                 double2* __restrict__ out) {
    const int b = blockIdx.x;        // batch element (uniform -> s_load params)
    const int e = threadIdx.x;       // 0..255 -> flattened (row,col) of 16x16
    const int r = e >> 4, c = e & 15;
    const int i = r >> 2, k = r & 3, j = c >> 2, l = c & 3;

    // Uniform scalar parameter loads (address depends only on blockIdx)
    const double Om = (double)Omega[b];
    const double De = (double)Delta[b];
    const double dp = (double)dph[b];
    const double D1 = De + (double)dd1[b] + dp;
    const double D2 = De + (double)dd2[b] + dp;
    const double V  = (double)Vv[b];
    const double hOm = 0.5 * Om;

    // Im L[(i,k),(j,l)] = -H[i,j]*d_kl + d_ij*H[l,k]
    double im = 0.0;
    if (k == l) im -= Hval(i, j, D1, D2, V, hOm);
    if (i == j) im += Hval(l, k, D1, D2, V, hOm);

    // Re L = DECAY (batch-independent):
    //   diag: -0.5*Gamma*popcount(r)
    //   +Gamma at (r, r+10) for r in {0,1,4,5} and (r, r+5) for r in {0,2,8,10}
    double re = 0.0;
    if (r == c) {
        re = -0.5 * kGamma * (double)__popc(r);
    } else if ((c == r + 10) && ((r & 2) == 0) && (r < 8)) {
        re = kGamma;
    } else if ((c == r + 5) && ((r & 1) == 0) && ((r & 4) == 0)) {
        re = kGamma;
    }

    double2* gp = out + (((size_t)b << 8) + (size_t)e);

#if defined(__AMDGCN__) && __has_builtin(__builtin_amdgcn_global_store_async_from_lds_b128)
    // CDNA5 path: stage the 4KB tile in LDS, drain via the async LDS->global
    // store engine (tracked by ASYNCcnt) instead of holding data in VGPRs.
    __shared__ double2 tile[NTHREADS];
    tile[e].x = re;
    tile[e].y = im;
    __syncthreads();  // ensure ds_store complete before async engine reads LDS
    __builtin_amdgcn_global_store_async_from_lds_b128(
        (gv4i_ptr)gp,
        (lv4i_ptr)&tile[e],
        /*offset=*/0, /*cpol=*/0);
  #if __has_builtin(__builtin_amdgcn_s_wait_asynccnt)
    __builtin_amdgcn_s_wait_asynccnt(0);
  #else
    asm volatile("s_wait_asynccnt 0x0" ::: "memory");
  #endif
#else
    double2 v;
    v.x = re;
    v.y = im;
    *gp = v;  // coalesced global_store_b128
#endif
}

extern "C" void kernel_launch(void* const* d_in, const int* in_sizes, int n_in,
                              void* d_out, int out_size, void* d_ws, size_t ws_size,
                              hipStream_t stream) {
    const float* Omega = (const float*)d_in[0];
    const float* Delta = (const float*)d_in[1];
    const float* dd1   = (const float*)d_in[2];
    const float* dd2   = (const float*)d_in[3];
    const float* dph   = (const float*)d_in[4];
    const float* Vv    = (const float*)d_in[5];
    const int B = in_sizes[0];  // 65536

    lindblad_liouvillian_kernel<<<B, NTHREADS, 0, stream>>>(
        Omega, Delta, dd1, dd2, dph, Vv, (double2*)d_out);
}